// Model_77086073028793
// MI455X (gfx1250) — compile-verified
//
#include <hip/hip_runtime.h>
#include <math.h>

typedef float v2f __attribute__((ext_vector_type(2)));
typedef float v8f __attribute__((ext_vector_type(8)));

__device__ __forceinline__ unsigned fkey(float f) {
    unsigned b = __float_as_uint(f);
    return (b & 0x80000000u) ? ~b : (b | 0x80000000u);
}
__device__ __forceinline__ float fdec(unsigned k) {
    unsigned b = (k & 0x80000000u) ? (k & 0x7FFFFFFFu) : ~k;
    return __uint_as_float(b);
}

// ---- k_prep: u_l = wcl@fcw, u_r = wcr@fcw, cb = bc.fcw + fcb  (1 block, 128 thr)
__global__ void k_prep(const float* __restrict__ wcl, const float* __restrict__ wcr,
                       const float* __restrict__ fcw, const float* __restrict__ bc,
                       const float* __restrict__ fcb,
                       float* __restrict__ ul, float* __restrict__ ur, float* __restrict__ cb) {
    int t = threadIdx.x;  // 0..127
    float sl = 0.f, sr = 0.f;
    for (int j = 0; j < 128; ++j) {
        float f = fcw[j];
        sl += wcl[t * 128 + j] * f;
        sr += wcr[t * 128 + j] * f;
    }
    ul[t] = sl; ur[t] = sr;
    if (t == 0) {
        float s = fcb[0];
        for (int j = 0; j < 128; ++j) s += bc[j] * fcw[j];
        *cb = s;
    }
}

// ---- k_edge1: degree + 7-wide layer-1 aggregation (thread per edge)
__global__ void k_edge1(const int* __restrict__ ei, const float* __restrict__ x,
                        float* __restrict__ cnt, float* __restrict__ agg1, int E) {
    int e = blockIdx.x * blockDim.x + threadIdx.x;
    if (e >= E) return;
    int s = ei[e], d = ei[E + e];
    unsafeAtomicAdd(&cnt[d], 1.0f);
    const float* xs = x + (size_t)s * 7;
    float* ad = agg1 + (size_t)d * 8;
#pragma unroll
    for (int k = 0; k < 7; ++k) unsafeAtomicAdd(&ad[k], xs[k]);
}

// ---- k_inv: inv = 1/max(cnt,1)
__global__ void k_inv(const float* __restrict__ cnt, float* __restrict__ inv, int N) {
    int i = blockIdx.x * blockDim.x + threadIdx.x;
    if (i < N) inv[i] = 1.0f / fmaxf(cnt[i], 1.0f);
}

// ---- k_h1: h1 = tanh([agg1*inv || x] @ [[w1l];[w1r]] + b1)  (block of 128 per node)
__global__ void k_h1(const float* __restrict__ x, const float* __restrict__ agg1,
                     const float* __restrict__ inv, const float* __restrict__ w1l,
                     const float* __restrict__ w1r, const float* __restrict__ b1,
                     float* __restrict__ h1) {
    __shared__ float sIn[14];
    int i = blockIdx.x;
    int u = threadIdx.x;  // 0..127
    if (u < 7) sIn[u] = agg1[(size_t)i * 8 + u] * inv[i];
    else if (u < 14) sIn[u] = x[(size_t)i * 7 + (u - 7)];
    __syncthreads();
    float acc = b1[u];
#pragma unroll
    for (int k = 0; k < 7; ++k) acc += sIn[k] * w1l[k * 128 + u];
#pragma unroll
    for (int k = 0; k < 7; ++k) acc += sIn[7 + k] * w1r[k * 128 + u];
    h1[(size_t)i * 128 + u] = tanhf(acc);
}

// ---- k_agg2: big edge pass, 128-wide (one wave per edge, float4 per lane)
__global__ void k_agg2(const int* __restrict__ ei, const float* __restrict__ h1,
                       float* __restrict__ agg2, int E) {
    int gid = blockIdx.x * blockDim.x + threadIdx.x;
    int e = gid >> 5;
    int lane = threadIdx.x & 31;
    if (e >= E) return;
    int s = ei[e], d = ei[E + e];
    float4 v = ((const float4*)(h1 + (size_t)s * 128))[lane];
    float* a = agg2 + (size_t)d * 128 + lane * 4;
    unsafeAtomicAdd(a + 0, v.x);
    unsafeAtomicAdd(a + 1, v.y);
    unsafeAtomicAdd(a + 2, v.z);
    unsafeAtomicAdd(a + 3, v.w);
}

// ---- k_h2: WMMA f32 GEMM  h2 = tanh([agg2*inv || h1] @ [[w2l];[w2r]] + b2)
// block: 256 thr = 8 waves; block covers 16 nodes x 128 outputs; wave -> 16x16 tile
__global__ void k_h2(const float* __restrict__ agg2, const float* __restrict__ h1,
                     const float* __restrict__ inv, const float* __restrict__ w2l,
                     const float* __restrict__ w2r, const float* __restrict__ b2,
                     float* __restrict__ h2, int N) {
    __shared__ float sA[16 * 264];  // 16 nodes x 256 feats, padded row stride 264
    const int base = blockIdx.x * 16;
    const int tid = threadIdx.x;
    for (int idx = tid; idx < 16 * 256; idx += 256) {
        int row = idx >> 8, col = idx & 255;
        int node = base + row;
        float v = 0.f;
        if (node < N)
            v = (col < 128) ? agg2[(size_t)node * 128 + col] * inv[node]
                            : h1[(size_t)node * 128 + (col - 128)];
        sA[row * 264 + col] = v;
    }
    __syncthreads();
    const int lane = tid & 31;
    const int half = lane >> 4;   // 0/1
    const int m = lane & 15;
    const int col = (tid >> 5) * 16 + m;  // output column 0..127
    v8f acc = {0.f, 0.f, 0.f, 0.f, 0.f, 0.f, 0.f, 0.f};
    for (int k0 = 0; k0 < 256; k0 += 4) {
        v2f a, b;
        // A 16x4 f32 layout: lanes 0-15 -> K = k0+0 / k0+1 ; lanes 16-31 -> K = k0+2 / k0+3
        a.x = sA[m * 264 + k0 + 2 * half];
        a.y = sA[m * 264 + k0 + 2 * half + 1];
        // B 4x16 f32 layout (row striped across lanes): same K split, N = col
        const float* W = (k0 < 128) ? (w2l + k0 * 128) : (w2r + (k0 - 128) * 128);
        b.x = W[(2 * half) * 128 + col];
        b.y = W[(2 * half + 1) * 128 + col];
        acc = __builtin_amdgcn_wmma_f32_16x16x4_f32(false, a, false, b, (short)0, acc,
                                                    false, false);
    }
    float bias = b2[col];
#pragma unroll
    for (int r = 0; r < 8; ++r) {
        int node = base + r + 8 * half;  // C/D layout: vgpr r -> M=r (half0) / r+8 (half1)
        if (node < N) h2[(size_t)node * 128 + col] = tanhf(acc[r] + bias);
    }
}

// ---- k_scal: per-node scalars (wave per node): s_a=h2.wal, c_l=h2.ul, r_a=h2.war, r_c=h2.ur
__global__ void k_scal(const float* __restrict__ h2, const float* __restrict__ wal,
                       const float* __restrict__ war, const float* __restrict__ ul,
                       const float* __restrict__ ur, float2* __restrict__ scal2,
                       float2* __restrict__ rv2, int N) {
    int node = (blockIdx.x * blockDim.x + threadIdx.x) >> 5;
    int lane = threadIdx.x & 31;
    if (node >= N) return;
    float4 h = ((const float4*)(h2 + (size_t)node * 128))[lane];
    float4 wa = ((const float4*)wal)[lane];
    float4 wr = ((const float4*)war)[lane];
    float4 vl = ((const float4*)ul)[lane];
    float4 vr = ((const float4*)ur)[lane];
    float p0 = h.x * wa.x + h.y * wa.y + h.z * wa.z + h.w * wa.w;
    float p1 = h.x * vl.x + h.y * vl.y + h.z * vl.z + h.w * vl.w;
    float p2 = h.x * wr.x + h.y * wr.y + h.z * wr.z + h.w * wr.w;
    float p3 = h.x * vr.x + h.y * vr.y + h.z * vr.z + h.w * vr.w;
    for (int off = 16; off > 0; off >>= 1) {
        p0 += __shfl_down(p0, off, 32);
        p1 += __shfl_down(p1, off, 32);
        p2 += __shfl_down(p2, off, 32);
        p3 += __shfl_down(p3, off, 32);
    }
    if (lane == 0) {
        scal2[node] = make_float2(p0, p1);
        rv2[node] = make_float2(p2, p3);
    }
}

// ---- k_aggscal: scalar edge pass for both heads (thread per edge)
__global__ void k_aggscal(const int* __restrict__ ei, const float2* __restrict__ scal2,
                          float* __restrict__ aggscal, int E) {
    int e = blockIdx.x * blockDim.x + threadIdx.x;
    if (e >= E) return;
    int s = ei[e], d = ei[E + e];
    float2 v = scal2[s];
    unsafeAtomicAdd(&aggscal[2 * (size_t)d + 0], v.x);
    unsafeAtomicAdd(&aggscal[2 * (size_t)d + 1], v.y);
}

// ---- k_heads: actor logits (masked) -> d_out, global max key; critic node value -> batch pool
__global__ void k_heads(const float* __restrict__ x, const float* __restrict__ inv,
                        const float2* __restrict__ aggs, const float2* __restrict__ rv,
                        const float* __restrict__ ba, const float* __restrict__ cb,
                        const int* __restrict__ batch, float* __restrict__ aout,
                        float* __restrict__ bsum, float* __restrict__ bcnt,
                        unsigned* __restrict__ gmax, int N) {
    __shared__ unsigned smax;
    if (threadIdx.x == 0) smax = 0u;
    __syncthreads();
    int i = blockIdx.x * blockDim.x + threadIdx.x;
    if (i < N) {
        float iv = inv[i];
        float2 ag = aggs[i];
        float2 r = rv[i];
        float a = ag.x * iv + r.x + ba[0];
        if (x[(size_t)i * 7 + 3] != 0.f || x[(size_t)i * 7 + 4] != 0.f) a = -__builtin_inff();
        aout[i] = a;
        atomicMax(&smax, fkey(a));
        float v = ag.y * iv + r.y + cb[0];
        int b = batch[i];
        unsafeAtomicAdd(&bsum[b], v);
        unsafeAtomicAdd(&bcnt[b], 1.0f);
    }
    __syncthreads();
    if (threadIdx.x == 0) atomicMax(gmax, smax);
}

// ---- k_sumexp: sum of exp(a - max) (grid stride)
__global__ void k_sumexp(const float* __restrict__ aout, const unsigned* __restrict__ gmax,
                         float* __restrict__ gsum, int N) {
    float maxv = fdec(*gmax);
    float p = 0.f;
    for (int i = blockIdx.x * blockDim.x + threadIdx.x; i < N; i += gridDim.x * blockDim.x)
        p += expf(aout[i] - maxv);
    for (int off = 16; off > 0; off >>= 1) p += __shfl_down(p, off, 32);
    __shared__ float sw[8];
    if ((threadIdx.x & 31) == 0) sw[threadIdx.x >> 5] = p;
    __syncthreads();
    if (threadIdx.x < 32) {
        float v = (threadIdx.x < 8) ? sw[threadIdx.x] : 0.f;
        for (int off = 4; off > 0; off >>= 1) v += __shfl_down(v, off, 32);
        if (threadIdx.x == 0) unsafeAtomicAdd(gsum, v);
    }
}

// ---- k_afinal: a - max - log(sum)
__global__ void k_afinal(float* __restrict__ aout, const unsigned* __restrict__ gmax,
                         const float* __restrict__ gsum, int N) {
    int i = blockIdx.x * blockDim.x + threadIdx.x;
    if (i >= N) return;
    float c = fdec(*gmax) + logf(*gsum);
    aout[i] = aout[i] - c;
}

// ---- k_cfinal: tanh(batch mean)
__global__ void k_cfinal(const float* __restrict__ bsum, const float* __restrict__ bcnt,
                         float* __restrict__ out, int B) {
    int b = threadIdx.x;
    if (b < B) out[b] = tanhf(bsum[b] / fmaxf(bcnt[b], 1.0f));
}

extern "C" void kernel_launch(void* const* d_in, const int* in_sizes, int n_in,
                              void* d_out, int out_size, void* d_ws, size_t ws_size,
                              hipStream_t stream) {
    const float* x    = (const float*)d_in[0];
    const int*   ei   = (const int*)d_in[1];
    const int*   batc = (const int*)d_in[2];
    const float* w1l  = (const float*)d_in[3];
    const float* b1   = (const float*)d_in[4];
    const float* w1r  = (const float*)d_in[5];
    const float* w2l  = (const float*)d_in[6];
    const float* b2   = (const float*)d_in[7];
    const float* w2r  = (const float*)d_in[8];
    const float* wal  = (const float*)d_in[9];
    const float* ba   = (const float*)d_in[10];
    const float* war  = (const float*)d_in[11];
    const float* wcl  = (const float*)d_in[12];
    const float* bc   = (const float*)d_in[13];
    const float* wcr  = (const float*)d_in[14];
    const float* fcw  = (const float*)d_in[15];
    const float* fcb  = (const float*)d_in[16];
    float* out = (float*)d_out;

    const int N = in_sizes[0] / 7;
    const int E = in_sizes[1] / 2;
    const int B = out_size - N;
    const size_t Ns = (size_t)N;

    float* ws = (float*)d_ws;
    float*  inv     = ws;                 // N
    float*  cnt     = ws + Ns;            // N
    float*  agg1    = ws + 2 * Ns;        // 8N
    float*  h1      = ws + 10 * Ns;       // 128N
    float*  agg2    = ws + 138 * Ns;      // 128N
    float*  h2      = ws + 266 * Ns;      // 128N
    float2* scal2   = (float2*)(ws + 10 * Ns);  // alias dead h1 (used post-k_h2)
    float2* rv2     = (float2*)(ws + 12 * Ns);
    float*  aggscal = ws + 14 * Ns;             // 2N, alias dead h1
    float*  small   = ws + 394 * Ns;
    float*  ul   = small;            // 128
    float*  ur   = small + 128;      // 128
    float*  cb   = small + 256;      // 1
    float*  gsum = small + 257;      // 1
    unsigned* gmax = (unsigned*)(small + 260);
    float*  bsum = small + 288;      // <=64
    float*  bcnt = small + 352;      // <=64

    // zero accumulation regions (capture-safe)
    hipMemsetAsync(cnt, 0, Ns * sizeof(float), stream);
    hipMemsetAsync(agg1, 0, 8 * Ns * sizeof(float), stream);
    hipMemsetAsync(agg2, 0, 128 * Ns * sizeof(float), stream);
    hipMemsetAsync(small, 0, 512 * sizeof(float), stream);

    k_prep<<<1, 128, 0, stream>>>(wcl, wcr, fcw, bc, fcb, ul, ur, cb);
    k_edge1<<<(E + 255) / 256, 256, 0, stream>>>(ei, x, cnt, agg1, E);
    k_inv<<<(N + 255) / 256, 256, 0, stream>>>(cnt, inv, N);
    k_h1<<<N, 128, 0, stream>>>(x, agg1, inv, w1l, w1r, b1, h1);
    k_agg2<<<(E + 7) / 8, 256, 0, stream>>>(ei, h1, agg2, E);
    k_h2<<<(N + 15) / 16, 256, 0, stream>>>(agg2, h1, inv, w2l, w2r, b2, h2, N);
    hipMemsetAsync(aggscal, 0, 2 * Ns * sizeof(float), stream);  // after h1 is dead
    k_scal<<<(N + 7) / 8, 256, 0, stream>>>(h2, wal, war, ul, ur, scal2, rv2, N);
    k_aggscal<<<(E + 255) / 256, 256, 0, stream>>>(ei, scal2, aggscal, E);
    k_heads<<<(N + 255) / 256, 256, 0, stream>>>(x, inv, (const float2*)aggscal,
                                                 (const float2*)rv2, ba, cb, batc,
                                                 out, bsum, bcnt, gmax, N);
    k_sumexp<<<1024, 256, 0, stream>>>(out, gmax, gsum, N);
    k_afinal<<<(N + 255) / 256, 256, 0, stream>>>(out, gmax, gsum, N);
    k_cfinal<<<1, 64, 0, stream>>>(bsum, bcnt, out + N, B);
    (void)n_in; (void)ws_size;
}